// DifferentiableSimulator_23364622090931
// MI455X (gfx1250) — compile-verified
//
#include <hip/hip_runtime.h>
#include <hip/hip_bf16.h>

#define DT 0.01f

typedef float v2f __attribute__((ext_vector_type(2)));
typedef float v8f __attribute__((ext_vector_type(8)));

// Affine map on state s=(x,v):  x' = x + q*v + cx ; v' = p*v + cv
// Stored as float4 (p, q, cx, cv). amap_comp(F,G) = "apply F first, then G".
__device__ __forceinline__ float4 amap_comp(float4 F, float4 G) {
  float4 R;
  R.x = G.x * F.x;                 // p  = pG*pF
  R.y = F.y + G.y * F.x;           // q  = qF + qG*pF
  R.z = F.z + G.y * F.w + G.z;     // cx = cxF + qG*cvF + cxG
  R.w = G.x * F.w + G.w;           // cv = pG*cvF + cvG
  return R;
}

// ---------------- Phase 1: per-chunk carry (scan from zero state) --------------
__global__ void __launch_bounds__(256)
k1_chunk_carry(const float* __restrict__ actions,
               const float* __restrict__ massp,
               const float* __restrict__ fricp,
               float2* __restrict__ carry, int L) {
  int c = blockIdx.x * blockDim.x + threadIdx.x;
  float m     = fabsf(*massp) + 0.001f;
  float beta  = DT / m;
  float alpha = 1.0f - (*fricp) * DT / m;
  const float* u = actions + (size_t)c * (size_t)L;
  float x = 0.0f, v = 0.0f;
  for (int k = 0; k < L; k += 4) {
    if ((k & 127) == 0) __builtin_prefetch(u + k + 256, 0, 0); // global_prefetch_b8
    float4 uu = *reinterpret_cast<const float4*>(u + k);
    v = fmaf(alpha, v, beta * uu.x); x = fmaf(DT, v, x);
    v = fmaf(alpha, v, beta * uu.y); x = fmaf(DT, v, x);
    v = fmaf(alpha, v, beta * uu.z); x = fmaf(DT, v, x);
    v = fmaf(alpha, v, beta * uu.w); x = fmaf(DT, v, x);
  }
  carry[c] = make_float2(x, v);
}

// ---------------- Phase 2: scan of chunk carries (one workgroup) ---------------
__global__ void __launch_bounds__(1024)
k2_scan(const float2* __restrict__ carry,
        float2* __restrict__ starts,
        const float* __restrict__ init,
        const float* __restrict__ massp,
        const float* __restrict__ fricp,
        float2* __restrict__ out, int L, int nchunk) {
  __shared__ float4 sm[1024];
  int t = threadIdx.x;
  float m     = fabsf(*massp) + 0.001f;
  float oma   = (*fricp) * DT / m;      // 1 - alpha (exact)
  float alpha = 1.0f - oma;
  float pL = powf(alpha, (float)L);
  float qL = DT * alpha * (1.0f - pL) / oma;

  int K  = nchunk / 1024;
  int c0 = t * K;
  float4 G = make_float4(1.f, 0.f, 0.f, 0.f);
  for (int j = 0; j < K; ++j) {
    float2 cc = carry[c0 + j];
    G = amap_comp(G, make_float4(pL, qL, cc.x, cc.y));
  }
  sm[t] = G;
  __syncthreads();
  for (int d = 1; d < 1024; d <<= 1) {
    float4 left = make_float4(1.f, 0.f, 0.f, 0.f);
    if (t >= d) left = sm[t - d];
    float4 self = sm[t];
    __syncthreads();
    sm[t] = amap_comp(left, self);
    __syncthreads();
  }
  float4 E = make_float4(1.f, 0.f, 0.f, 0.f);
  if (t > 0) E = sm[t - 1];              // exclusive prefix map

  float x0 = init[0], v0 = init[1];
  if (t == 0) out[0] = make_float2(x0, v0);   // row 0 = initial state
  for (int j = 0; j < K; ++j) {
    int c = c0 + j;
    float xs = x0 + E.y * v0 + E.z;
    float vs = E.x * v0 + E.w;
    starts[c] = make_float2(xs, vs);
    float2 cc = carry[c];
    E = amap_comp(E, make_float4(pL, qL, cc.x, cc.y));
  }
}

// ---------------- Phase 3: WMMA tensor-scan emit -------------------------------
// One wave handles 16 chunks (the 16 N-columns of a 16x16 WMMA tile).
// Per 16-step tile:  V = Lmat@U + pvec*v_in ;  X = Mx@U + (x_in + DT*Sp*v_in)
// B operands are software-pipelined (next tile loaded during current tile's
// WMMA chain) so the wave doesn't stall on s_wait_loadcnt before every WMMA.
__global__ void __launch_bounds__(256)
k3_emit(const float* __restrict__ actions,
        const float2* __restrict__ starts,
        const float* __restrict__ massp,
        const float* __restrict__ fricp,
        float2* __restrict__ out, int L) {
  int gtid = blockIdx.x * blockDim.x + threadIdx.x;
  int wave = gtid >> 5;
  int lane = threadIdx.x & 31;
  int hi   = lane >> 4;        // which half of the wave
  int n    = lane & 15;        // column (chunk within wave) / A-row index
  int col  = wave * 16 + n;    // global chunk id for this lane's column

  float m     = fabsf(*massp) + 0.001f;
  float oma   = (*fricp) * DT / m;     // 1 - alpha
  float alpha = 1.0f - oma;
  float beta  = DT / m;
  float la    = logf(alpha);
  auto apow = [&](int e) -> float { return expf(la * (float)e); }; // alpha^e

  // A operands (f32 16x4 layout: lanes 0-15 hold K = j0,j0+1 in vgpr0/1 with
  // j0 = 4*kk; lanes 16-31 hold K = j0+2, j0+3).
  v2f AL[4], AX[4];
  int M = n; // A-matrix row for this lane
  #pragma unroll
  for (int kk = 0; kk < 4; ++kk) {
    int j0 = 4 * kk + 2 * hi;
    float a0 = 0.f, a1 = 0.f, b0 = 0.f, b1 = 0.f;
    if (j0     <= M) { a0 = beta * apow(M - j0);
                       b0 = DT * beta * (1.0f - apow(M - j0 + 1)) / oma; }
    if (j0 + 1 <= M) { a1 = beta * apow(M - j0 - 1);
                       b1 = DT * beta * (1.0f - apow(M - j0)) / oma; }
    AL[kk].x = a0; AL[kk].y = a1;
    AX[kk].x = b0; AX[kk].y = b1;
  }
  // Per-lane accumulator-seed coefficients (C layout: vgpr r -> row i = r+8*hi)
  float pv[8], sp[8];
  #pragma unroll
  for (int r = 0; r < 8; ++r) {
    int i = r + 8 * hi;
    pv[r] = apow(i + 1);                          // coeff of v_in in v_out[i]
    sp[r] = DT * alpha * (1.0f - pv[r]) / oma;    // coeff of v_in in x_out[i]
  }

  float2 st = starts[col];
  float xs = st.x, vs = st.y;
  const float* ub = actions + (size_t)col * (size_t)L;
  size_t orow0 = (size_t)col * (size_t)L + 1 + 8 * (size_t)hi;
  int ntiles = L >> 4;

  // Prologue: load tile 0's B operand
  // (f32 4x16 layout mirrors A: vgpr0 = {K0 | K2}, vgpr1 = {K1 | K3})
  v2f Bc[4];
  #pragma unroll
  for (int kk = 0; kk < 4; ++kk) {
    Bc[kk] = *reinterpret_cast<const v2f*>(ub + 4 * kk + 2 * hi);
  }

  for (int tIdx = 0; tIdx < ntiles; ++tIdx) {
    // Issue next tile's B loads first — they retire during this tile's WMMAs.
    // Last iteration re-loads the current tile (stays in bounds, result unused).
    int nxt = (tIdx + 1 < ntiles) ? (tIdx + 1) : tIdx;
    v2f Bn[4];
    #pragma unroll
    for (int kk = 0; kk < 4; ++kk) {
      Bn[kk] = *reinterpret_cast<const v2f*>(ub + nxt * 16 + 4 * kk + 2 * hi);
    }

    // Seed accumulators with carry contribution
    v8f V, X;
    #pragma unroll
    for (int r = 0; r < 8; ++r) {
      V[r] = pv[r] * vs;
      X[r] = fmaf(sp[r], vs, xs);
    }
    // 8 chained f32 WMMAs: K=16 in 4 steps of 4, for both V and X (B in regs)
    #pragma unroll
    for (int kk = 0; kk < 4; ++kk) {
      V = __builtin_amdgcn_wmma_f32_16x16x4_f32(false, AL[kk], false, Bc[kk],
                                                (short)0, V, false, false);
      X = __builtin_amdgcn_wmma_f32_16x16x4_f32(false, AX[kk], false, Bc[kk],
                                                (short)0, X, false, false);
    }
    // Carry = row 15 of each result: vgpr7 on lanes 16..31 (lane 16+n = col n)
    float vlast = V[7], xlast = X[7];
    vs = __shfl(vlast, 16 + n, 32);
    xs = __shfl(xlast, 16 + n, 32);
    // Store interleaved (x, v) rows for this tile
    size_t rb = orow0 + (size_t)tIdx * 16;
    #pragma unroll
    for (int r = 0; r < 8; ++r) {
      out[rb + r] = make_float2(X[r], V[r]);
    }
    // Rotate double buffer
    #pragma unroll
    for (int kk = 0; kk < 4; ++kk) Bc[kk] = Bn[kk];
  }
}

// ---------------- Launcher -----------------------------------------------------
extern "C" void kernel_launch(void* const* d_in, const int* in_sizes, int n_in,
                              void* d_out, int out_size, void* d_ws, size_t ws_size,
                              hipStream_t stream) {
  const float* init    = (const float*)d_in[0];
  const float* actions = (const float*)d_in[1];
  const float* massp   = (const float*)d_in[2];
  const float* fricp   = (const float*)d_in[3];
  long long T = in_sizes[1];          // 8388608 = 2^23

  int nchunk = 32768;                 // L = 256, 16 tiles of 16 steps per chunk
  while (nchunk > 1024 && (T % ((long long)nchunk * 16)) != 0) nchunk >>= 1;
  int L = (int)(T / nchunk);

  float2* carry  = (float2*)d_ws;           // nchunk float2
  float2* starts = carry + nchunk;          // nchunk float2  (total 512 KB)
  float2* out    = (float2*)d_out;

  k1_chunk_carry<<<nchunk / 256, 256, 0, stream>>>(actions, massp, fricp, carry, L);
  k2_scan<<<1, 1024, 0, stream>>>(carry, starts, init, massp, fricp, out, L, nchunk);
  k3_emit<<<nchunk / 128, 256, 0, stream>>>(actions, starts, massp, fricp, out, L);
}